// QuanvolutionFilterQuantum_65481071399648
// MI455X (gfx1250) — compile-verified
//
#include <hip/hip_runtime.h>

typedef __attribute__((ext_vector_type(2))) float v2f;
typedef __attribute__((ext_vector_type(8))) float v8f;

// ---------------------------------------------------------------------------
// Kernel 1: build the fixed 16x16 circuit matrix C from params.
// state indexing: s = q0*8 + q1*4 + q2*2 + q3  (qubit i <-> bit (3-i))
// Gates applied left-to-right onto C (C = G * C), matching the reference:
//   per layer: RY(params[l][i]) on qubit i (i=0..3), then CNOT(0,1),(1,2),(2,3),(3,0)
// ---------------------------------------------------------------------------
__global__ void quanv_build_circuit(const float* __restrict__ params, int n_layers,
                                    float* __restrict__ Cmat) {
  if (threadIdx.x != 0 || blockIdx.x != 0) return;
  float C[16][16];
  for (int s = 0; s < 16; ++s)
    for (int k = 0; k < 16; ++k)
      C[s][k] = (s == k) ? 1.0f : 0.0f;

  for (int l = 0; l < n_layers; ++l) {
    // RY on each qubit: row-pair rotation  b0 = c*a0 - s*a1 ; b1 = s*a0 + c*a1
    for (int i = 0; i < 4; ++i) {
      float th = params[l * 4 + i] * 0.5f;
      float cc = __cosf(th), ss = __sinf(th);
      int bit = 8 >> i;
      for (int s = 0; s < 16; ++s) {
        if (s & bit) continue;
        int s1 = s | bit;
        for (int k = 0; k < 16; ++k) {
          float a0 = C[s][k], a1 = C[s1][k];
          C[s][k]  = cc * a0 - ss * a1;
          C[s1][k] = ss * a0 + cc * a1;
        }
      }
    }
    // CNOT chain: swap rows (ctrl=1,tgt=0) <-> (ctrl=1,tgt=1)
    const int ctrls[4] = {0, 1, 2, 3};
    const int tgts[4]  = {1, 2, 3, 0};
    for (int e = 0; e < 4; ++e) {
      int cb = 8 >> ctrls[e], tb = 8 >> tgts[e];
      for (int s = 0; s < 16; ++s) {
        if ((s & cb) && !(s & tb)) {
          int s1 = s | tb;
          for (int k = 0; k < 16; ++k) {
            float t = C[s][k]; C[s][k] = C[s1][k]; C[s1][k] = t;
          }
        }
      }
    }
  }
  for (int s = 0; s < 16; ++s)
    for (int k = 0; k < 16; ++k)
      Cmat[s * 16 + k] = C[s][k];
}

// ---------------------------------------------------------------------------
// Kernel 2: per wave, process tiles of 16 patches with V_WMMA_F32_16X16X4_F32.
//   Y(16 states x 16 patches) = C(16x16) * Psi(16x16), K split into 4 chunks.
// A (16x4 f32) layout: VGPR0: lanes0-15 (M=lane, K=0), lanes16-31 (M=lane-16, K=2)
//                      VGPR1: K=1 / K=3     -> A[c] = {C[m][k0], C[m][k0+1]}
// B (4x16 f32) layout mirrors A:  B = {psi[k0], psi[k0+1]} with k0 = 4c + 2*half
// D (16x16 f32): VGPR r: lanes0-15 = (state r, patch lane),
//                        lanes16-31 = (state r+8, patch lane-16)
// ---------------------------------------------------------------------------
__global__ __launch_bounds__(256) void quanv_wmma(
    const float* __restrict__ x, const float* __restrict__ Cmat,
    float* __restrict__ out, int nTiles, int nPatches) {
  const int lane = threadIdx.x & 31;
  const int half = (lane >= 16) ? 1 : 0;
  const int m = lane & 15;
  const int waveId = blockIdx.x * (blockDim.x >> 5) + (threadIdx.x >> 5);
  const int nWaves = gridDim.x * (blockDim.x >> 5);

  // Load the fixed A-operand (circuit matrix) once per wave.
  v2f A[4];
#pragma unroll
  for (int c = 0; c < 4; ++c) {
    int k0 = 4 * c + 2 * half;
    A[c].x = Cmat[m * 16 + k0];
    A[c].y = Cmat[m * 16 + k0 + 1];
  }

  for (int tile = waveId; tile < nTiles; tile += nWaves) {
    int g = tile * 16 + m;                    // global patch index (this lane)
    int gp = (g < nPatches) ? g : (nPatches - 1);
    int b  = gp / 196;
    int p  = gp - b * 196;
    int pr = p / 14;
    int pc = p - pr * 14;
    const float* img = x + (size_t)b * 784 + (size_t)(2 * pr) * 28 + 2 * pc;
    float a00 = img[0],  a01 = img[1];
    float a10 = img[28], a11 = img[29];

    // Encoding RYs: psi[s] = prod_i (bit ? sin : cos)(theta_i / 2)
    float c0 = __cosf(a00 * 0.5f), s0 = __sinf(a00 * 0.5f);
    float c1 = __cosf(a01 * 0.5f), s1 = __sinf(a01 * 0.5f);
    float c2 = __cosf(a10 * 0.5f), s2 = __sinf(a10 * 0.5f);
    float c3 = __cosf(a11 * 0.5f), s3 = __sinf(a11 * 0.5f);
    float psi[16];
#pragma unroll
    for (int i = 0; i < 16; ++i) {
      float f0 = (i & 8) ? s0 : c0;
      float f1 = (i & 4) ? s1 : c1;
      float f2 = (i & 2) ? s2 : c2;
      float f3 = (i & 1) ? s3 : c3;
      psi[i] = (f0 * f1) * (f2 * f3);
    }

    // Y = C * Psi via 4 chained f32 WMMAs (K = 16 in chunks of 4).
    v8f acc = {};
#pragma unroll
    for (int c = 0; c < 4; ++c) {
      v2f Bv;
      Bv.x = half ? psi[4 * c + 2] : psi[4 * c + 0];
      Bv.y = half ? psi[4 * c + 3] : psi[4 * c + 1];
      acc = __builtin_amdgcn_wmma_f32_16x16x4_f32(
          /*neg_a=*/false, A[c], /*neg_b=*/false, Bv,
          /*c_mod=*/(short)0, acc, /*reuse_a=*/false, /*reuse_b=*/false);
    }

    // Probabilities and signed marginals. State s = r (lanes<16) or r+8.
    // qubit0 sign = bit3 (lane-half dependent); qubits 1..3 = bits 2..0 of r.
    float P[8];
#pragma unroll
    for (int r = 0; r < 8; ++r) { float v = acc[r]; P[r] = v * v; }
    float t = 0.f, z1 = 0.f, z2 = 0.f, z3 = 0.f;
#pragma unroll
    for (int r = 0; r < 8; ++r) {
      t  += P[r];
      z1 += (r & 4) ? -P[r] : P[r];
      z2 += (r & 2) ? -P[r] : P[r];
      z3 += (r & 1) ? -P[r] : P[r];
    }
    float z0 = half ? -t : t;
    // combine lane-halves (states 0-7 + states 8-15 for the same patch)
    z0 += __shfl_xor(z0, 16, 32);
    z1 += __shfl_xor(z1, 16, 32);
    z2 += __shfl_xor(z2, 16, 32);
    z3 += __shfl_xor(z3, 16, 32);

    if (!half && g < nPatches) {
      float4 o = make_float4(z0, z1, z2, z3);
      *(float4*)(out + (size_t)g * 4) = o;   // out[b][p*4 + j] == out[g*4 + j]
    }
  }
}

extern "C" void kernel_launch(void* const* d_in, const int* in_sizes, int n_in,
                              void* d_out, int out_size, void* d_ws, size_t ws_size,
                              hipStream_t stream) {
  const float* x      = (const float*)d_in[0];
  const float* params = (const float*)d_in[1];
  float* out  = (float*)d_out;
  float* Cmat = (float*)d_ws;                 // 256 floats of scratch

  int n_layers = in_sizes[1] / 4;             // params shape (L, 4)
  int B        = in_sizes[0] / 784;           // images
  int nPatches = B * 196;
  int nTiles   = (nPatches + 15) / 16;

  quanv_build_circuit<<<1, 32, 0, stream>>>(params, n_layers, Cmat);

  // ~8 tiles per wave, 8 waves (256 threads) per block
  int wavesNeeded = (nTiles + 7) / 8;
  int blocks = (wavesNeeded + 7) / 8;
  if (blocks < 1) blocks = 1;
  quanv_wmma<<<blocks, 256, 0, stream>>>(x, Cmat, out, nTiles, nPatches);
}